// CylindricalPointEncoder_71176198029924
// MI455X (gfx1250) — compile-verified
//
#include <hip/hip_runtime.h>
#include <hip/hip_bf16.h>
#include <math.h>

typedef __attribute__((ext_vector_type(2))) float v2f;
typedef __attribute__((ext_vector_type(8))) float v8f;

#define NBINS (48*48*48)   // 110592 voxel keys
#define EPS   1e-5f

// ---------------------------------------------------------------- fills
__global__ void fill_i32_kernel(int* __restrict__ p, int v, int n) {
  int i = blockIdx.x * blockDim.x + threadIdx.x;
  if (i < n) p[i] = v;
}
__global__ void fill_f32_kernel(float* __restrict__ p, float v, int n) {
  int i = blockIdx.x * blockDim.x + threadIdx.x;
  if (i < n) p[i] = v;
}

// ---------------------------------------------------------------- unique(coords) via counting sort over 48^3 bins
__global__ void key_hist_kernel(const int* __restrict__ coords, int* __restrict__ keys,
                                int* __restrict__ hist, int n) {
  int i = blockIdx.x * blockDim.x + threadIdx.x;
  if (i < n) {
    int x = coords[3*i+0], y = coords[3*i+1], z = coords[3*i+2];
    int k = (x * 48 + y) * 48 + z;          // lexicographic key
    keys[i] = k;
    atomicAdd(&hist[k], 1);                 // int atomic: deterministic
  }
}

__global__ void scan_block_kernel(const int* __restrict__ hist, int* __restrict__ rank,
                                  int* __restrict__ bsums, int nbins) {
  __shared__ int sh[1024];
  int tid = threadIdx.x;
  int gid = blockIdx.x * 1024 + tid;
  int flag = (gid < nbins && hist[gid] > 0) ? 1 : 0;
  sh[tid] = flag;
  __syncthreads();
  for (int off = 1; off < 1024; off <<= 1) {   // Hillis-Steele inclusive scan
    int t = (tid >= off) ? sh[tid - off] : 0;
    __syncthreads();
    sh[tid] += t;
    __syncthreads();
  }
  if (gid < nbins) rank[gid] = sh[tid] - flag;  // exclusive
  if (tid == 1023) bsums[blockIdx.x] = sh[1023];
}

__global__ void scan_sums_kernel(int* __restrict__ bsums, int nb) {
  if (blockIdx.x == 0 && threadIdx.x == 0) {
    int acc = 0;
    for (int i = 0; i < nb; ++i) { int t = bsums[i]; bsums[i] = acc; acc += t; }
  }
}

__global__ void add_offsets_kernel(int* __restrict__ rank, const int* __restrict__ bsums, int nbins) {
  int gid = blockIdx.x * 1024 + threadIdx.x;
  if (gid < nbins) rank[gid] += bsums[blockIdx.x];
}

__global__ void emit_unique_kernel(const int* __restrict__ hist, const int* __restrict__ rank,
                                   int* __restrict__ unq, int* __restrict__ cnt_rank, int nbins) {
  int k = blockIdx.x * blockDim.x + threadIdx.x;
  if (k < nbins && hist[k] > 0) {
    int u = rank[k];
    int x = k / 2304, rem = k % 2304;
    unq[3*u+0] = x; unq[3*u+1] = rem / 48; unq[3*u+2] = rem % 48;
    cnt_rank[u] = hist[k];
  }
}

__global__ void map_inv_kernel(const int* __restrict__ keys, const int* __restrict__ rank,
                               int* __restrict__ inv, int n) {
  int i = blockIdx.x * blockDim.x + threadIdx.x;
  if (i < n) inv[i] = rank[keys[i]];
}

// ---------------------------------------------------------------- deterministic per-column sum/sumsq
// small-D version (feats, D=9): one block per column
template <int D>
__global__ void col_stats_kernel(const float* __restrict__ y, int n, float* __restrict__ out) {
  __shared__ float s1[256], s2[256];
  const int col = blockIdx.x;
  float s = 0.f, q = 0.f;
  for (int r = threadIdx.x; r < n; r += 256) {
    float v = y[(long)r * D + col];
    s += v; q += v * v;
  }
  s1[threadIdx.x] = s; s2[threadIdx.x] = q;
  __syncthreads();
  for (int off = 128; off > 0; off >>= 1) {
    if (threadIdx.x < off) {
      s1[threadIdx.x] += s1[threadIdx.x + off];
      s2[threadIdx.x] += s2[threadIdx.x + off];
    }
    __syncthreads();
  }
  if (threadIdx.x == 0) { out[col] = s1[0]; out[D + col] = s2[0]; }
}

// coalesced partial version for D in {64,128,256}: threads cover D consecutive cols
template <int D>
__global__ void col_partial_kernel(const float* __restrict__ y, int n, float* __restrict__ part) {
  constexpr int RPT = 256 / D;                 // rows in flight per pass
  __shared__ float s1[256], s2[256];
  const int tid  = threadIdx.x;
  const int col  = tid % D;
  const int rsub = tid / D;
  float s = 0.f, q = 0.f;
  for (int r = blockIdx.x * RPT + rsub; r < n; r += gridDim.x * RPT) {
    float v = y[(long)r * D + col];            // fully coalesced
    s += v; q += v * v;
  }
  s1[tid] = s; s2[tid] = q;
  __syncthreads();
  if (tid < D) {
    float as = s1[tid], aq = s2[tid];
#pragma unroll
    for (int j = 1; j < RPT; ++j) { as += s1[j * D + tid]; aq += s2[j * D + tid]; }
    part[(long)blockIdx.x * 2 * D + tid]     = as;   // fixed order => deterministic
    part[(long)blockIdx.x * 2 * D + D + tid] = aq;
  }
}

__global__ void stats_reduce_kernel(const float* __restrict__ part, int G, int twoD,
                                    float* __restrict__ stats) {
  int i = blockIdx.x * blockDim.x + threadIdx.x;
  if (i < twoD) {
    float s = 0.f;
    for (int b = 0; b < G; ++b) s += part[(long)b * twoD + i];  // fixed order
    stats[i] = s;
  }
}

__global__ void bn_finalize_kernel(const float* __restrict__ stats, const float* __restrict__ g,
                                   const float* __restrict__ b, int d, int n,
                                   float* __restrict__ scale, float* __restrict__ shift) {
  int i = blockIdx.x * blockDim.x + threadIdx.x;
  if (i < d) {
    float inv_n = 1.f / (float)n;
    float mean  = stats[i] * inv_n;
    float var   = stats[d + i] * inv_n - mean * mean;   // biased var (training-mode BN)
    float rstd  = rsqrtf(var + EPS);
    float sc    = g[i] * rstd;
    scale[i] = sc;
    shift[i] = b[i] - mean * sc;
  }
}

// ---------------------------------------------------------------- weight packing into WMMA B-fragment order
// Wp[nt][kk][lane] = { W[k0][col], W[k0+1][col] },  col = nt*16 + lane%16,  k0 = kk*4 + (lane>>4)*2
template <int FIN, int FOUT>
__global__ void pack_w_kernel(const float* __restrict__ W, float* __restrict__ Wp) {
  constexpr int KS = (FIN + 3) / 4;
  constexpr int NT = FOUT / 16;
  int idx = blockIdx.x * blockDim.x + threadIdx.x;      // over NT*KS*32
  if (idx < NT * KS * 32) {
    int lane = idx & 31;
    int t    = idx >> 5;
    int kk   = t % KS;
    int nt   = t / KS;
    int col  = nt * 16 + (lane & 15);
    int k0   = kk * 4 + (lane >> 4) * 2;
    Wp[2*idx+0] = (k0 + 0 < FIN) ? W[(long)(k0 + 0) * FOUT + col] : 0.f;
    Wp[2*idx+1] = (k0 + 1 < FIN) ? W[(long)(k0 + 1) * FOUT + col] : 0.f;
  }
}

// ---------------------------------------------------------------- fused (pre-op -> GEMM+bias -> post-relu) via fp32 WMMA
// B tiles stream into LDS with async-to-LDS copies, double-buffered across output tiles.
// PRE: 0 = BN, 1 = BN+ReLU, 2 = zero rows with cnt==0 (segment-max padding mask)
template <int FIN, int FOUT, int PRE, bool POSTRELU>
__global__ void mlp_stage_kernel(const float* __restrict__ xin,
                                 const float* __restrict__ scale,
                                 const float* __restrict__ shift,
                                 const int*   __restrict__ cnt,
                                 const float* __restrict__ Wp,     // packed fragments
                                 const float* __restrict__ bias,   // [FOUT]
                                 float* __restrict__ yout, int n) {
  constexpr int KS = (FIN + 3) / 4;              // K-steps of 4
  constexpr int NT = FOUT / 16;                  // 16-column output tiles
  constexpr int TILE_BYTES = KS * 64 * 4;        // one packed tile
  constexpr int TILE_XFERS = TILE_BYTES / 16;    // b128 transfers per tile

  __shared__ __align__(16) float smem[2][KS * 64];

  const int lane = threadIdx.x & 31;
  const int wave = threadIdx.x >> 5;
  const int rowBase = blockIdx.x * 128 + wave * 16;
  const int m  = lane & 15;                      // A: row in tile; B/C: column in tile
  const int hi = lane >> 4;
  const int row = rowBase + m;
  const int rclamp = (row < n) ? row : (n - 1);  // clamp loads; stores guarded wave-uniformly

  // async copy of one packed weight tile into LDS buffer `buf`
  auto issue_tile = [&](int nt, int buf) {
    const uint64_t src = (uint64_t)(uintptr_t)Wp + (uint64_t)nt * TILE_BYTES;
    const unsigned dbase = (unsigned)(uintptr_t)(&smem[buf][0]);
    for (int t = threadIdx.x; t < TILE_XFERS; t += 256) {
      unsigned d = dbase + t * 16;
      uint64_t g = src + (uint64_t)t * 16;
      asm volatile("global_load_async_to_lds_b128 %0, %1, off"
                   :: "v"(d), "v"(g) : "memory");
    }
  };

  // ---- A fragments for this wave's 16 rows, pre-op applied, kept in VGPRs
  float a[2 * KS];
  const float mask = (PRE == 2) ? ((cnt != nullptr && cnt[rclamp] > 0) ? 1.f : 0.f) : 1.f;
  if (FIN % 4 == 0) {
    const float* xr = xin + (long)rclamp * FIN;
#pragma unroll
    for (int kk = 0; kk < KS; ++kk) {
      const int k0 = kk * 4 + hi * 2;
      v2f t = *(const v2f*)(xr + k0);            // 8B-aligned b64 load
#pragma unroll
      for (int j = 0; j < 2; ++j) {
        float v = t[j];
        if (PRE == 0 || PRE == 1) {
          v = v * scale[k0 + j] + shift[k0 + j];
          if (PRE == 1) v = fmaxf(v, 0.f);
        } else {
          v = v * mask;
        }
        a[kk * 2 + j] = v;
      }
    }
  } else {
#pragma unroll
    for (int kk = 0; kk < KS; ++kk) {
      const int k0 = kk * 4 + hi * 2;
#pragma unroll
      for (int j = 0; j < 2; ++j) {
        const int k = k0 + j;
        float v = 0.f;
        if (k < FIN) {
          v = xin[(long)rclamp * FIN + k];
          if (PRE == 0 || PRE == 1) {
            v = v * scale[k] + shift[k];
            if (PRE == 1) v = fmaxf(v, 0.f);
          } else {
            v = v * mask;
          }
        }
        a[kk * 2 + j] = v;
      }
    }
  }

  // prologue: stream tile 0
  issue_tile(0, 0);
  asm volatile("s_wait_asynccnt 0x0" ::: "memory");
  __syncthreads();

  // wave-uniform store guard, forced into an SGPR -> pure scalar branch
  const int fullTile = __builtin_amdgcn_readfirstlane((rowBase + 16 <= n) ? 1 : 0);

  for (int nt = 0; nt < NT; ++nt) {
    const int cur = nt & 1;
    if (nt + 1 < NT) issue_tile(nt + 1, cur ^ 1);   // prefetch next tile

    const int col = nt * 16 + m;
    const float bcol = bias[col];
    v8f c;
#pragma unroll
    for (int r = 0; r < 8; ++r) c[r] = bcol;     // seed accumulator with bias

    const v2f* bsrc = (const v2f*)&smem[cur][0];
#pragma unroll
    for (int kk = 0; kk < KS; ++kk) {
      v2f av; av.x = a[kk * 2 + 0]; av.y = a[kk * 2 + 1];
      v2f bv = bsrc[kk * 32 + lane];             // ds_load_b64, conflict-free
      c = __builtin_amdgcn_wmma_f32_16x16x4_f32(
              /*neg_a=*/false, av, /*neg_b=*/false, bv,
              /*c_mod=*/(short)0, c, /*reuse_a=*/false, /*reuse_b=*/false);
    }

    // lane holds column `col`; VGPR r holds row hi*8 + r
    float* ybase = yout + (long)(rowBase + hi * 8) * FOUT + col;
    if (fullTile) {
#pragma unroll
      for (int r = 0; r < 8; ++r) {
        float v = c[r];
        if (POSTRELU) v = fmaxf(v, 0.f);
        ybase[(long)r * FOUT] = v;
      }
    } else {
#pragma unroll
      for (int r = 0; r < 8; ++r) {
        if (rowBase + hi * 8 + r < n) {
          float v = c[r];
          if (POSTRELU) v = fmaxf(v, 0.f);
          ybase[(long)r * FOUT] = v;
        }
      }
    }

    if (nt + 1 < NT) {                                   // block-uniform
      asm volatile("s_wait_asynccnt 0x0" ::: "memory");  // next tile landed
      __syncthreads();                                   // all waves done with smem[cur]
    }
  }
}

// ---------------------------------------------------------------- segment-max scatter (order-independent => deterministic)
__global__ void scatter_max_kernel(const float* __restrict__ y4, const int* __restrict__ inv,
                                   float* __restrict__ pooled, int n) {
  int idx = blockIdx.x * blockDim.x + threadIdx.x;
  if (idx < n * 64) {
    int i = idx >> 6, c = idx & 63;
    float v = y4[idx];
    float* addr = pooled + ((long)inv[i] << 6) + c;
    // IEEE float max via integer atomics
    if (v >= 0.f) atomicMax((int*)addr, __float_as_int(v));
    else          atomicMin((unsigned int*)addr, __float_as_uint(v));
  }
}

// ---------------------------------------------------------------- launcher
extern "C" void kernel_launch(void* const* d_in, const int* in_sizes, int n_in,
                              void* d_out, int out_size, void* d_ws, size_t ws_size,
                              hipStream_t stream) {
  const float* feats  = (const float*)d_in[0];
  const int*   coords = (const int*)d_in[1];
  const float* g0 = (const float*)d_in[2];  const float* be0 = (const float*)d_in[3];
  const float* g1 = (const float*)d_in[4];  const float* be1 = (const float*)d_in[5];
  const float* g2 = (const float*)d_in[6];  const float* be2 = (const float*)d_in[7];
  const float* g3 = (const float*)d_in[8];  const float* be3 = (const float*)d_in[9];
  const float* w1 = (const float*)d_in[10]; const float* b1 = (const float*)d_in[11];
  const float* w2 = (const float*)d_in[12]; const float* b2 = (const float*)d_in[13];
  const float* w3 = (const float*)d_in[14]; const float* b3 = (const float*)d_in[15];
  const float* w4 = (const float*)d_in[16]; const float* b4 = (const float*)d_in[17];
  const float* wp = (const float*)d_in[18]; const float* bp = (const float*)d_in[19];

  const int N = in_sizes[0] / 9;
  const int G = 256;                     // partial-stats blocks

  // workspace carve-out
  char* ws = (char*)d_ws;
  size_t off = 0;
  auto carve = [&](size_t bytes) -> char* {
    char* p = ws + off;
    off = (off + bytes + 255) & ~(size_t)255;
    return p;
  };
  float* y1       = (float*)carve((size_t)N * 64  * 4);
  float* y2       = (float*)carve((size_t)N * 128 * 4);
  float* y3       = (float*)carve((size_t)N * 256 * 4);
  float* y4       = (float*)carve((size_t)N * 64  * 4);
  float* pooled   = (float*)carve((size_t)N * 64  * 4);
  int*   keys     = (int*)  carve((size_t)N * 4);
  int*   inv      = (int*)  carve((size_t)N * 4);
  int*   cnt_rank = (int*)  carve((size_t)N * 4);
  int*   hist     = (int*)  carve((size_t)NBINS * 4);
  int*   rank     = (int*)  carve((size_t)NBINS * 4);
  int*   bsums    = (int*)  carve(1024 * 4);
  float* stats    = (float*)carve(512 * 4);
  float* scale    = (float*)carve(256 * 4);
  float* shift    = (float*)carve(256 * 4);
  float* part     = (float*)carve((size_t)G * 512 * 4);
  float* w1p      = (float*)carve((size_t)4  * 3  * 64 * 4);   // NT*KS*64 floats
  float* w2p      = (float*)carve((size_t)8  * 16 * 64 * 4);
  float* w3p      = (float*)carve((size_t)16 * 32 * 64 * 4);
  float* w4p      = (float*)carve((size_t)4  * 64 * 64 * 4);
  float* wpp      = (float*)carve((size_t)1  * 16 * 64 * 4);

  float* out_pooled = (float*)d_out;                          // [N,16] fp32
  int*   out_unq    = (int*)(out_pooled + (size_t)N * 16);    // [N,3] int32

  const int T = 256;
  const int gN    = (N + T - 1) / T;
  const int gBins = (NBINS + T - 1) / T;
  const int gScan = (NBINS + 1023) / 1024;                    // 108
  const int gRows = (N + 127) / 128;                          // 128 rows / block (8 waves x 16)

  // ---- pack weights into WMMA fragment order
  pack_w_kernel<9,   64 ><<<(4*3*32   + T - 1) / T, T, 0, stream>>>(w1, w1p);
  pack_w_kernel<64,  128><<<(8*16*32  + T - 1) / T, T, 0, stream>>>(w2, w2p);
  pack_w_kernel<128, 256><<<(16*32*32 + T - 1) / T, T, 0, stream>>>(w3, w3p);
  pack_w_kernel<256, 64 ><<<(4*64*32  + T - 1) / T, T, 0, stream>>>(w4, w4p);
  pack_w_kernel<64,  16 ><<<(1*16*32  + T - 1) / T, T, 0, stream>>>(wp, wpp);

  // ---- unique / inverse / counts (counting sort over 48^3 keys)
  fill_i32_kernel<<<gBins, T, 0, stream>>>(hist, 0, NBINS);
  key_hist_kernel<<<gN, T, 0, stream>>>(coords, keys, hist, N);
  scan_block_kernel<<<gScan, 1024, 0, stream>>>(hist, rank, bsums, NBINS);
  scan_sums_kernel<<<1, 32, 0, stream>>>(bsums, gScan);
  add_offsets_kernel<<<gScan, 1024, 0, stream>>>(rank, bsums, NBINS);
  fill_i32_kernel<<<(3 * N + T - 1) / T, T, 0, stream>>>(out_unq, -1, 3 * N);
  fill_i32_kernel<<<gN, T, 0, stream>>>(cnt_rank, 0, N);
  emit_unique_kernel<<<gBins, T, 0, stream>>>(hist, rank, out_unq, cnt_rank, NBINS);
  map_inv_kernel<<<gN, T, 0, stream>>>(keys, rank, inv, N);

  // ---- stage 1: BN0(feats) @ W1 + b1
  col_stats_kernel<9><<<9, T, 0, stream>>>(feats, N, stats);
  bn_finalize_kernel<<<1, T, 0, stream>>>(stats, g0, be0, 9, N, scale, shift);
  mlp_stage_kernel<9, 64, 0, false><<<gRows, T, 0, stream>>>(feats, scale, shift, nullptr, w1p, b1, y1, N);

  // ---- stage 2: relu(BN1(y1)) @ W2 + b2
  col_partial_kernel<64><<<G, T, 0, stream>>>(y1, N, part);
  stats_reduce_kernel<<<1, T, 0, stream>>>(part, G, 128, stats);
  bn_finalize_kernel<<<1, T, 0, stream>>>(stats, g1, be1, 64, N, scale, shift);
  mlp_stage_kernel<64, 128, 1, false><<<gRows, T, 0, stream>>>(y1, scale, shift, nullptr, w2p, b2, y2, N);

  // ---- stage 3: relu(BN2(y2)) @ W3 + b3
  col_partial_kernel<128><<<G, T, 0, stream>>>(y2, N, part);
  stats_reduce_kernel<<<1, T, 0, stream>>>(part, G, 256, stats);
  bn_finalize_kernel<<<1, T, 0, stream>>>(stats, g2, be2, 128, N, scale, shift);
  mlp_stage_kernel<128, 256, 1, false><<<gRows, T, 0, stream>>>(y2, scale, shift, nullptr, w3p, b3, y3, N);

  // ---- stage 4: relu(BN3(y3)) @ W4 + b4
  col_partial_kernel<256><<<G, T, 0, stream>>>(y3, N, part);
  stats_reduce_kernel<<<(512 + T - 1) / T, T, 0, stream>>>(part, G, 512, stats);
  bn_finalize_kernel<<<1, T, 0, stream>>>(stats, g3, be3, 256, N, scale, shift);
  mlp_stage_kernel<256, 64, 1, false><<<gRows, T, 0, stream>>>(y3, scale, shift, nullptr, w4p, b4, y4, N);

  // ---- segment-max pooling
  fill_f32_kernel<<<(N * 64 + T - 1) / T, T, 0, stream>>>(pooled, -INFINITY, N * 64);
  scatter_max_kernel<<<(N * 64 + T - 1) / T, T, 0, stream>>>(y4, inv, pooled, N);

  // ---- projection: relu(mask(pooled) @ Wp + bp)  -> d_out (padding rows emit relu(bp))
  mlp_stage_kernel<64, 16, 2, true><<<gRows, T, 0, stream>>>(pooled, nullptr, nullptr, cnt_rank, wpp, bp, out_pooled, N);
}